// JointBoneDirEncoder_28355374088636
// MI455X (gfx1250) — compile-verified
//
#include <hip/hip_runtime.h>
#include <cstdint>

#define NJ 24
#define SPB 8                 // samples per block
#define TPB (SPB * NJ)        // 192 threads = 6 wave32

// ---- CDNA5 builtin availability probes (compile-time) ----
#if defined(__has_builtin)
#  if __has_builtin(__builtin_amdgcn_global_load_async_to_lds_b128)
#    define HAVE_ASYNC_LDS 1
#  endif
#  if __has_builtin(__builtin_amdgcn_s_wait_asynccnt)
#    define HAVE_WAIT_ASYNC 1
#  endif
#  if __has_builtin(__builtin_amdgcn_global_prefetch)
#    define HAVE_GLOBAL_PREFETCH 1
#  endif
#endif
#ifndef HAVE_ASYNC_LDS
#  define HAVE_ASYNC_LDS 0
#endif
#ifndef HAVE_WAIT_ASYNC
#  define HAVE_WAIT_ASYNC 0
#endif
#ifndef HAVE_GLOBAL_PREFETCH
#  define HAVE_GLOBAL_PREFETCH 0
#endif

typedef int v4i __attribute__((vector_size(16)));
typedef __attribute__((address_space(1))) v4i* gv4i_p;
typedef __attribute__((address_space(3))) v4i* lv4i_p;
typedef __attribute__((address_space(1))) void* gvoid_p;

__device__ __forceinline__ void prefetch_wgp(const void* p) {
#if HAVE_GLOBAL_PREFETCH
  // cpol 0 -> TH=RT, scope=0 (WGP): pull into all cache levels on miss
  __builtin_amdgcn_global_prefetch((gvoid_p)(uintptr_t)p, 0);
#else
  __builtin_prefetch(p, 0, 3);   // global_prefetch_b8 (SE scope)
#endif
}

// 24 entries of 5 bits packed into two u64 (entries 0..11 in lo, 12..23 in hi)
__device__ __forceinline__ int unpack5(uint64_t lo, uint64_t hi, int j) {
  uint64_t w = (j < 12) ? (lo >> (5 * j)) : (hi >> (5 * (j - 12)));
  return (int)(w & 31ull);
}

__global__ __launch_bounds__(TPB) void jbde_kernel(
    const float* __restrict__ bones,   // (N, 24, 3)
    const float* __restrict__ kps,     // (N, 24, 3)
    const float* __restrict__ skts,    // (N, 24, 4, 4)
    float* __restrict__ out,           // (N, 24, 6)
    int N)
{
  // PARENT_IDXS / CHILD_IDXS / CHILD_MASK baked as immediates
  constexpr uint64_t P_LO = 0x041CC520C4100000ULL;
  constexpr uint64_t P_HI = 0x0AD2728C1CD62529ULL;
  constexpr uint64_t C_LO = 0x05A92B5250731480ULL;
  constexpr uint64_t C_HI = 0x0BDAF6AD2727C60FULL;
  constexpr unsigned MASKBITS = 0x3F71FEu;   // bit j == CHILD_MASK[j]

  __shared__ __align__(16) float kps_s[SPB * NJ * 3];   // 576 floats = 2304 B

  const int t  = threadIdx.x;
  const int n0 = blockIdx.x * SPB;
  const int ns = (N - n0 < SPB) ? (N - n0) : SPB;       // valid samples this block

  const int nl = t / NJ;                 // local sample 0..7
  const int j  = t - nl * NJ;            // joint 0..23
  const size_t idx = (size_t)(n0 + nl) * NJ + j;
  const bool active = (nl < ns);

  // Prefetch this thread's streaming data into all cache levels while we
  // stage kps into LDS.
  if (active) {
    prefetch_wgp((const void*)(skts + idx * 16));
    prefetch_wgp((const void*)(bones + idx * 3));
  }

  // ---- Stage this block's kps (ns*72 floats) into LDS, coalesced b128 ----
  const float* kps_blk = kps + (size_t)n0 * (NJ * 3);
  const int nchunks = ns * 18;           // float4 chunks (288 B per sample)
#if HAVE_ASYNC_LDS
  if (t < nchunks) {
    // generic->AS1/AS3 via integer cast (AS0->AS3 lowers to 32-bit truncation)
    gv4i_p g = (gv4i_p)(uintptr_t)(kps_blk + t * 4);
    lv4i_p l = (lv4i_p)(uint32_t)(uintptr_t)(kps_s + t * 4);
    __builtin_amdgcn_global_load_async_to_lds_b128(g, l, 0, 0);
  }
#  if HAVE_WAIT_ASYNC
  __builtin_amdgcn_s_wait_asynccnt(0);
#  else
  asm volatile("s_wait_asynccnt 0" ::: "memory");
#  endif
#else
  if (t < nchunks) {
    float4 v = ((const float4*)kps_blk)[t];
    ((float4*)kps_s)[t] = v;
  }
#endif
  __syncthreads();

  if (!active) return;

  // ---- Load skts rows 0..2 (row 3 never contributes to out[...,:3]) ----
  const float4* M = ((const float4*)skts) + idx * 4;
  float4 m0 = M[0];
  float4 m1 = M[1];
  float4 m2 = M[2];

  const int   pj = unpack5(P_LO, P_HI, j);
  const int   cj = unpack5(C_LO, C_HI, j);
  const float* kp = kps_s + nl * (NJ * 3);

  float px = kp[pj * 3 + 0], py = kp[pj * 3 + 1], pz = kp[pj * 3 + 2];
  float cx = kp[cj * 3 + 0], cy = kp[cj * 3 + 1], cz = kp[cj * 3 + 2];

  // vec_parent = normalize(M[0:3] * [p,1])
  float vpx = fmaf(m0.x, px, fmaf(m0.y, py, fmaf(m0.z, pz, m0.w)));
  float vpy = fmaf(m1.x, px, fmaf(m1.y, py, fmaf(m1.z, pz, m1.w)));
  float vpz = fmaf(m2.x, px, fmaf(m2.y, py, fmaf(m2.z, pz, m2.w)));
  float np  = sqrtf(fmaf(vpx, vpx, fmaf(vpy, vpy, vpz * vpz)));
  float rp  = 1.0f / fmaxf(np, 1e-12f);
  vpx *= rp; vpy *= rp; vpz *= rp;

  // vec_child = normalize(M[0:3] * [c,1])
  float vcx = fmaf(m0.x, cx, fmaf(m0.y, cy, fmaf(m0.z, cz, m0.w)));
  float vcy = fmaf(m1.x, cx, fmaf(m1.y, cy, fmaf(m1.z, cz, m1.w)));
  float vcz = fmaf(m2.x, cx, fmaf(m2.y, cy, fmaf(m2.z, cz, m2.w)));
  float nc  = sqrtf(fmaf(vcx, vcx, fmaf(vcy, vcy, vcz * vcz)));
  float rc  = 1.0f / fmaxf(nc, 1e-12f);
  vcx *= rc; vcy *= rc; vcz *= rc;

  // Joints without exactly one child: replace with Rodrigues first column.
  if (((MASKBITS >> j) & 1u) == 0u) {
    const float* b = bones + idx * 3;
    float bx = b[0], by = b[1], bz = b[2];
    float tt  = fmaf(bx, bx, fmaf(by, by, bz * bz)) + 1e-8f;
    float th  = sqrtf(tt);
    float inv = 1.0f / th;
    float kx = bx * inv, ky = by * inv, kz = bz * inv;
    float c  = __cosf(th);
    float s  = __sinf(th);
    float omc = 1.0f - c;
    float okx = omc * kx;
    vcx = fmaf(okx, kx, c);
    vcy = fmaf(okx, ky, s * kz);
    vcz = fmaf(okx, kz, -s * ky);
  }

  // out[idx*6 .. idx*6+5]; byte offset idx*24 is 8-aligned -> three b64 stores
  float2* o = (float2*)(out + idx * 6);
  o[0] = make_float2(vpx, vpy);
  o[1] = make_float2(vpz, vcx);
  o[2] = make_float2(vcy, vcz);
}

extern "C" void kernel_launch(void* const* d_in, const int* in_sizes, int n_in,
                              void* d_out, int out_size, void* d_ws, size_t ws_size,
                              hipStream_t stream) {
  const float* bones = (const float*)d_in[0];
  const float* kps   = (const float*)d_in[1];
  const float* skts  = (const float*)d_in[2];
  float* out = (float*)d_out;

  const int N = in_sizes[0] / (NJ * 3);        // bones is (N, 24, 3)
  const int blocks = (N + SPB - 1) / SPB;
  jbde_kernel<<<blocks, TPB, 0, stream>>>(bones, kps, skts, out, N);
}